// GTShapelet_38723425141160
// MI455X (gfx1250) — compile-verified
//
#include <hip/hip_runtime.h>
#include <hip/hip_bf16.h>

// ---------------------------------------------------------------------------
// Problem constants (match reference)
// ---------------------------------------------------------------------------
#define BB 32
#define NN 1024
#define BN (BB * NN)          // 32768 nodes
#define DD 128
#define D2 256
#define SS 1025               // N + 1 (cls token appended)
#define HH 4
#define HD 32

typedef __attribute__((ext_vector_type(2))) float v2f;
typedef __attribute__((ext_vector_type(8))) float v8f;

__device__ __forceinline__ float gelu_exact(float x) {
    return 0.5f * x * (1.0f + erff(x * 0.70710678118654752440f));
}

// ---------------------------------------------------------------------------
// 1) Embedding gather: h0[i,:] = embed_table[node_ids[i], :]   (float4 wide)
// ---------------------------------------------------------------------------
__global__ void gather_embed(const int* __restrict__ ids,
                             const float4* __restrict__ embed,
                             float4* __restrict__ out) {
    int idx = blockIdx.x * blockDim.x + threadIdx.x;   // over BN * (DD/4)
    int n = idx >> 5;            // DD/4 == 32
    int c = idx & 31;
    out[idx] = embed[ids[n] * 32 + c];
}

// ---------------------------------------------------------------------------
// 2) Edge scatter: msg[dst[e], d] += ew[e] * h[src[e], d]
//    One thread per (edge, feature). Dh = 1<<SHIFT (128 or 256).
// ---------------------------------------------------------------------------
template <int SHIFT>
__global__ void scatter_edges(const float* __restrict__ h,
                              const int* __restrict__ src,
                              const int* __restrict__ dst,
                              const float* __restrict__ ew,
                              float* __restrict__ msg,
                              int nEdges) {
    long long idx = (long long)blockIdx.x * blockDim.x + threadIdx.x;
    int e = (int)(idx >> SHIFT);
    if (e >= nEdges) return;
    int d = (int)(idx & ((1 << SHIFT) - 1));
    int s = src[e], t = dst[e];
    float val = ew[e] * h[((long long)s << SHIFT) + d];
    unsafeAtomicAdd(&msg[((long long)t << SHIFT) + d], val);
}

// ---------------------------------------------------------------------------
// 3) WMMA f32 GIN GEMM:  OUT = gelu( (A + MSG) @ W + bias )
//    One wave computes a 16x16 tile of OUT via V_WMMA_F32_16X16X4_F32.
//    Compile-time K / LDW / NC -> fully unrolled K loop, immediate offsets.
//    A,MSG: [M,K] row-major; W: [K,LDW] row-major; OUT: [M,NC] row-major.
// ---------------------------------------------------------------------------
template <int K, int LDW, int NC, bool GELU>
__global__ void gemm_gin_wmma(const float* __restrict__ A,
                              const float* __restrict__ MSG,
                              const float* __restrict__ W,
                              const float* __restrict__ bias,
                              float* __restrict__ OUT) {
    int lane = threadIdx.x;            // 0..31, full wave (EXEC all ones)
    int m0 = blockIdx.x * 16;
    int n0 = blockIdx.y * 16;
    int ml = lane & 15;
    int hi = lane >> 4;                // 0: K pair {0,1}, 1: K pair {2,3}
    int kk = hi * 2;
    int col = n0 + ml;

    const float* arow = A   + (long long)(m0 + ml) * K + kk;
    const float* mrow = MSG + (long long)(m0 + ml) * K + kk;
    const float* wcol = W   + (long long)kk * LDW + col;

    v8f acc = {};
    #pragma unroll
    for (int k0 = 0; k0 < K; k0 += 4) {
        v2f av, bv;
        av.x = arow[k0]     + mrow[k0];
        av.y = arow[k0 + 1] + mrow[k0 + 1];
        bv.x = wcol[(long long)k0 * LDW];
        bv.y = wcol[(long long)(k0 + 1) * LDW];
        acc = __builtin_amdgcn_wmma_f32_16x16x4_f32(
            /*neg_a=*/false, av, /*neg_b=*/false, bv,
            /*c_mod=*/(short)0, acc, /*reuse_a=*/false, /*reuse_b=*/false);
    }

    float bc = bias[col];
    int rbase = m0 + hi * 8;           // C layout: VGPR r -> M=r (lo) / M=r+8 (hi)
    #pragma unroll
    for (int r = 0; r < 8; ++r) {
        float x = acc[r] + bc;
        if (GELU) x = gelu_exact(x);
        OUT[(long long)(rbase + r) * NC + col] = x;
    }
}

// ---------------------------------------------------------------------------
// 4) KV projection GEMM (WMMA): rows come from h3 (s<N) or cls row (s==N).
//    M = B*S = 32800 (divisible by 16), K=128, NC=256; Wkv/bkv pre-offset by
//    +D columns of in_proj_w/in_proj_b (ldw=384).
//    KV row layout: [k(128) | v(128)].
// ---------------------------------------------------------------------------
__global__ void kv_gemm_wmma(const float* __restrict__ H3,
                             const float* __restrict__ cls,
                             const float* __restrict__ Wkv,
                             const float* __restrict__ bkv,
                             float* __restrict__ KV) {
    int lane = threadIdx.x;
    int m0 = blockIdx.x * 16;
    int n0 = blockIdx.y * 16;
    int ml = lane & 15;
    int hi = lane >> 4;
    int kk = hi * 2;
    int col = n0 + ml;

    int m = m0 + ml;
    int b = m / SS;
    int s = m - b * SS;
    const float* arow = ((s < NN) ? (H3 + (long long)(b * NN + s) * DD) : cls) + kk;
    const float* wcol = Wkv + kk * 384 + col;

    v8f acc = {};
    #pragma unroll
    for (int k0 = 0; k0 < DD; k0 += 4) {
        v2f av, bv;
        av.x = arow[k0];
        av.y = arow[k0 + 1];
        bv.x = wcol[k0 * 384];
        bv.y = wcol[(k0 + 1) * 384];
        acc = __builtin_amdgcn_wmma_f32_16x16x4_f32(
            false, av, false, bv, (short)0, acc, false, false);
    }
    float bc = bkv[col];
    int rbase = m0 + hi * 8;
    #pragma unroll
    for (int r = 0; r < 8; ++r)
        KV[(long long)(rbase + r) * D2 + col] = acc[r] + bc;
}

// ---------------------------------------------------------------------------
// 5) q_cls = cls_embedding @ in_proj_w[:, 0:D] + in_proj_b[0:D]
//    (cls row of h is identical for all batches -> a single 128-vector)
// ---------------------------------------------------------------------------
__global__ void qcls_kernel(const float* __restrict__ cls,
                            const float* __restrict__ Wqkv,
                            const float* __restrict__ bqkv,
                            float* __restrict__ qcls) {
    int d = threadIdx.x;               // 128
    float acc = bqkv[d];
    #pragma unroll 8
    for (int j = 0; j < DD; ++j) acc += cls[j] * Wqkv[j * 384 + d];
    qcls[d] = acc;
}

// ---------------------------------------------------------------------------
// 6) cls-query attention: per (b, head) block.
//    scores[s] = q_cls_h . k[b,s,h,:]/sqrt(hd) (+ mask), softmax over s,
//    ctx[b, h*hd + j] = sum_s p[s] * v[b,s,h,j]
// ---------------------------------------------------------------------------
__global__ void attn_cls(const float* __restrict__ KV,
                         const float* __restrict__ qcls,
                         const unsigned char* __restrict__ pad_mask,
                         float* __restrict__ ctx) {
    __shared__ float sc[SS];
    __shared__ float red[256];
    int b = blockIdx.x >> 2;           // H == 4
    int h = blockIdx.x & 3;
    int tid = threadIdx.x;
    const float scale = 0.17677669529663688f;   // 1/sqrt(32)

    float q[HD];
    #pragma unroll
    for (int j = 0; j < HD; ++j) q[j] = qcls[h * HD + j];

    float lmax = -3.4e38f;
    for (int s = tid; s < SS; s += 256) {
        const float* k = KV + (long long)(b * SS + s) * D2 + h * HD;
        float d = 0.f;
        #pragma unroll
        for (int j = 0; j < HD; ++j) d += q[j] * k[j];
        d *= scale;
        if (s < NN && pad_mask[b * NN + s]) d -= 1e9f;
        sc[s] = d;
        lmax = fmaxf(lmax, d);
    }
    red[tid] = lmax;
    __syncthreads();
    for (int off = 128; off > 0; off >>= 1) {
        if (tid < off) red[tid] = fmaxf(red[tid], red[tid + off]);
        __syncthreads();
    }
    float m = red[0];
    __syncthreads();

    float lsum = 0.f;
    for (int s = tid; s < SS; s += 256) {
        float e = expf(sc[s] - m);
        sc[s] = e;
        lsum += e;
    }
    red[tid] = lsum;
    __syncthreads();
    for (int off = 128; off > 0; off >>= 1) {
        if (tid < off) red[tid] += red[tid + off];
        __syncthreads();
    }
    float inv = 1.f / red[0];
    __syncthreads();

    int hd = tid & 31;
    int chunk = tid >> 5;              // 0..7
    float acc = 0.f;
    for (int s = chunk; s < SS; s += 8)
        acc += sc[s] * KV[(long long)(b * SS + s) * D2 + DD + h * HD + hd];
    red[tid] = acc;
    __syncthreads();
    if (tid < 32) {
        float t = 0.f;
        #pragma unroll
        for (int c = 0; c < 8; ++c) t += red[c * 32 + hd];
        ctx[b * DD + h * HD + hd] = t * inv;
    }
}

// ---------------------------------------------------------------------------
// 7) out-projection (cls rows only) + residual + LayerNorm -> out[B, D]
// ---------------------------------------------------------------------------
__global__ void out_proj_ln(const float* __restrict__ ctx,
                            const float* __restrict__ Wo,
                            const float* __restrict__ bo,
                            const float* __restrict__ cls,
                            const float* __restrict__ g,
                            const float* __restrict__ beta,
                            float* __restrict__ out) {
    __shared__ float red[DD];
    int b = blockIdx.x, d = threadIdx.x;
    const float* c = ctx + b * DD;
    float o = bo[d];
    #pragma unroll 8
    for (int j = 0; j < DD; ++j) o += c[j] * Wo[j * DD + d];
    float y = cls[d] + o;

    red[d] = y; __syncthreads();
    for (int off = 64; off > 0; off >>= 1) {
        if (d < off) red[d] += red[d + off];
        __syncthreads();
    }
    float mu = red[0] * (1.f / DD);
    __syncthreads();
    float dv = y - mu;
    red[d] = dv * dv; __syncthreads();
    for (int off = 64; off > 0; off >>= 1) {
        if (d < off) red[d] += red[d + off];
        __syncthreads();
    }
    float var = red[0] * (1.f / DD);
    out[b * DD + d] = dv * rsqrtf(var + 1e-5f) * g[d] + beta[d];
}

// ---------------------------------------------------------------------------
// Launch
// ---------------------------------------------------------------------------
extern "C" void kernel_launch(void* const* d_in, const int* in_sizes, int n_in,
                              void* d_out, int out_size, void* d_ws, size_t ws_size,
                              hipStream_t stream) {
    const int*   node_ids = (const int*)d_in[0];
    const int*   src      = (const int*)d_in[1];
    const int*   dst      = (const int*)d_in[2];
    const unsigned char* pad_mask = (const unsigned char*)d_in[3];   // bool
    const float* ew       = (const float*)d_in[4];
    const float* embed    = (const float*)d_in[5];
    const float* W1 = (const float*)d_in[6];  const float* b1 = (const float*)d_in[7];
    const float* W2 = (const float*)d_in[8];  const float* b2 = (const float*)d_in[9];
    const float* W3 = (const float*)d_in[10]; const float* b3 = (const float*)d_in[11];
    const float* in_w = (const float*)d_in[12];
    const float* in_b = (const float*)d_in[13];
    const float* out_w = (const float*)d_in[14];
    const float* out_b = (const float*)d_in[15];
    const float* cls   = (const float*)d_in[16];
    const float* ln_g  = (const float*)d_in[17];
    const float* ln_b  = (const float*)d_in[18];
    float* out = (float*)d_out;

    // Workspace layout (floats)
    float* ws   = (float*)d_ws;
    float* bufA = ws;                                  // 32768*256
    float* bufB = bufA + (size_t)BN * D2;              // 32768*256
    float* msg  = bufB + (size_t)BN * D2;              // 32768*256
    float* kv   = msg  + (size_t)BN * D2;              // 32800*256
    float* qcls = kv   + (size_t)BB * SS * D2;         // 128
    float* ctx  = qcls + DD;                           // 32*128

    const int E = in_sizes[1];

    // 1) embedding gather -> bufA [BN, 128]
    gather_embed<<<(BN * (DD / 4)) / 256, 256, 0, stream>>>(
        node_ids, (const float4*)embed, (float4*)bufA);

    // ---- GIN layer 1: gelu((h0 + msg) @ W1 + b1) -> bufB [BN, 256] ----
    hipMemsetAsync(msg, 0, (size_t)BN * DD * sizeof(float), stream);
    scatter_edges<7><<<(unsigned)(((long long)E * DD) / 256), 256, 0, stream>>>(
        bufA, src, dst, ew, msg, E);
    gemm_gin_wmma<DD, D2, D2, true><<<dim3(BN / 16, D2 / 16), 32, 0, stream>>>(
        bufA, msg, W1, b1, bufB);

    // ---- GIN layer 2: gelu((h1 + msg) @ W2 + b2) -> bufA [BN, 256] ----
    hipMemsetAsync(msg, 0, (size_t)BN * D2 * sizeof(float), stream);
    scatter_edges<8><<<(unsigned)(((long long)E * D2) / 256), 256, 0, stream>>>(
        bufB, src, dst, ew, msg, E);
    gemm_gin_wmma<D2, D2, D2, true><<<dim3(BN / 16, D2 / 16), 32, 0, stream>>>(
        bufB, msg, W2, b2, bufA);

    // ---- GIN layer 3: gelu((h2 + msg) @ W3 + b3) -> bufB [BN, 128] ----
    hipMemsetAsync(msg, 0, (size_t)BN * D2 * sizeof(float), stream);
    scatter_edges<8><<<(unsigned)(((long long)E * D2) / 256), 256, 0, stream>>>(
        bufA, src, dst, ew, msg, E);
    gemm_gin_wmma<D2, DD, DD, true><<<dim3(BN / 16, DD / 16), 32, 0, stream>>>(
        bufA, msg, W3, b3, bufB);

    // ---- kv projection for all (b,s): KV [B*S, 256] ----
    kv_gemm_wmma<<<dim3((BB * SS) / 16, D2 / 16), 32, 0, stream>>>(
        bufB, cls, in_w + DD, in_b + DD, kv);

    // ---- q_cls (single 128-vector, identical across batches) ----
    qcls_kernel<<<1, DD, 0, stream>>>(cls, in_w, in_b, qcls);

    // ---- cls-query attention -> ctx [B, 128] ----
    attn_cls<<<BB * HH, 256, 0, stream>>>(kv, qcls, pad_mask, ctx);

    // ---- out projection + residual + LayerNorm -> out [B, 128] ----
    out_proj_ln<<<BB, DD, 0, stream>>>(ctx, out_w, out_b, cls, ln_g, ln_b, out);
}